// DecompLayer_32134945308821
// MI455X (gfx1250) — compile-verified
//
#include <hip/hip_runtime.h>

typedef float v2f __attribute__((ext_vector_type(2)));
typedef float v4f __attribute__((ext_vector_type(4)));
typedef float v8f __attribute__((ext_vector_type(8)));

#define B_   32
#define C_   256
#define H_   128
#define W_   128
#define R_   128
#define F_   256
#define HP_  126
#define WP_  126

#define TOR  8              // output rows per tile
#define TOC  16             // output cols per tile (== one WMMA N tile per row)
#define YR   (TOR + 2)      // 10 halo rows of y
#define YC   (TOC + 2)      // 18 halo cols of y
#define NP   (YR * YC)      // 180 y positions per rank
#define NPAD 192            // padded position count (12 N tiles)
#define NT1  12             // N-tiles in GEMM1
#define KC   32             // channels staged per LDS chunk

#define NXREG 24            // (KC*NPAD)/256 x-values per thread per chunk
#define NFREG 16            // (KC*R)/256 f3-values per thread per chunk

// K-minor staging buffers: [position][K], stride 34 (34*p mod 64 -> conflict-free b64)
#define KSTR 34
// r-minor buffers: [position][r], stride 132 (4*p + r -> all 64 banks on b64 reads)
#define RSTR 132

#define YS_FLOATS  (NP * RSTR)            // 23760   ys[p][r]
#define XS_FLOATS  (NPAD * KSTR)          // 6528    xs[p][k]
#define F3T_FLOATS (R_ * KSTR)            // 4352    f3t[r][k]
#define ZS_FLOATS  (TOR * TOC * RSTR)     // 16896   zs[op][r]
#define U_FLOATS   (ZS_FLOATS)            // union{xs+f3t (10880) , zs (16896)}
#define LDS_FLOATS (YS_FLOATS + U_FLOATS) // 40656 floats = 162,624 B

#define WMMA_F32(A, Bm, Cc) \
  __builtin_amdgcn_wmma_f32_16x16x4_f32(false, (A), false, (Bm), (short)0, (Cc), false, false)

__global__ __launch_bounds__(256)
void cp_conv_fused(const float* __restrict__ x,
                   const float* __restrict__ f0,   // [F, R]
                   const float* __restrict__ f1,   // [3, R]
                   const float* __restrict__ f2,   // [3, R]
                   const float* __restrict__ f3,   // [C, R]
                   float* __restrict__ out)        // [B, F, 126, 126]
{
  extern __shared__ float smem[];
  float* ys  = smem;                           // [NP][RSTR]
  float* xs  = smem + YS_FLOATS;               // [NPAD][KSTR]   (phase 1)
  float* f3t = smem + YS_FLOATS + XS_FLOATS;   // [R][KSTR]      (phase 1)
  float* zs  = smem + YS_FLOATS;               // [128 op][RSTR] (phases 2/3, aliases xs/f3t)

  const int tid  = threadIdx.x;
  const int wv   = tid >> 5;           // 8 waves
  const int lane = tid & 31;
  const int lm   = lane & 15;          // M/N index inside a 16x16 tile
  const int kb   = (lane >> 4) << 1;   // K base: lanes 0-15 -> K0/K1, lanes 16-31 -> K2/K3
  const int rowoff = (lane >> 4) << 3; // C/D layout: upper lane half holds M+8

  const int ow0 = blockIdx.x * TOC;
  const int oh0 = blockIdx.y * TOR;
  const int b   = blockIdx.z;

  // ================= Phase 1: y[r,p] = sum_c f3[c,r] * x[b,c, oh0+dr, ow0+dc]
  const int r0 = wv << 4;              // this wave's 16-rank slice
  v8f acc[NT1];
#pragma unroll
  for (int t = 0; t < NT1; ++t) acc[t] = (v8f){0.f,0.f,0.f,0.f,0.f,0.f,0.f,0.f};

  const float* xb = x + (size_t)b * C_ * H_ * W_;

  float xv[NXREG];                     // register double-buffer: next chunk of x
  float fv[NFREG];                     // register double-buffer: next chunk of f3

  // issue global loads for a chunk into registers (overlaps WMMA of prev chunk)
  auto load_chunk = [&](int c0) {
#pragma unroll
    for (int i = 0; i < NXREG; ++i) {
      int idx = tid + i * 256;
      int k = idx / NPAD;
      int p = idx - k * NPAD;          // consecutive tids -> consecutive p: coalesced
      float v = 0.f;
      if (p < NP) {
        int dr = p / YC;
        int dc = p - dr * YC;
        int gh = oh0 + dr;
        int gw = ow0 + dc;
        if (gh < H_ && gw < W_) {
          int o = ((c0 + k) * H_ + gh) * W_ + gw;
          v = xb[o];
          if (c0 + KC < C_)            // prefetch chunk after next (global_prefetch_b8)
            __builtin_prefetch(xb + o + KC * H_ * W_, 0, 1);
        }
      }
      xv[i] = v;
    }
#pragma unroll
    for (int i = 0; i < NFREG; ++i) {
      int idx = tid + i * 256;
      int k = idx >> 7;                // idx / 128
      int r = idx & 127;               // consecutive tids -> consecutive r: coalesced
      fv[i] = f3[(c0 + k) * R_ + r];
    }
  };

  load_chunk(0);

  for (int c0 = 0; c0 < C_; c0 += KC) {
    __syncthreads();                   // previous chunk's LDS reads complete
    // commit register buffers to LDS (K-minor)
#pragma unroll
    for (int i = 0; i < NXREG; ++i) {
      int idx = tid + i * 256;
      int k = idx / NPAD;
      int p = idx - k * NPAD;
      xs[p * KSTR + k] = xv[i];
    }
#pragma unroll
    for (int i = 0; i < NFREG; ++i) {
      int idx = tid + i * 256;
      int k = idx >> 7;
      int r = idx & 127;
      f3t[r * KSTR + k] = fv[i];
    }
    __syncthreads();

    if (c0 + KC < C_)
      load_chunk(c0 + KC);             // HBM latency hidden under the K-loop below

#pragma unroll 2
    for (int kk = 0; kk < KC; kk += 4) {
      // A[m,k] = f3[c0+kk+k, r0+m] : one b64 from transposed LDS copy
      const v2f a = *(const v2f*)(f3t + (r0 + lm) * KSTR + kk + kb);
#pragma unroll
      for (int t = 0; t < NT1; ++t) {
        // B[k,n] = x[c0+kk+k, p=16t+n] : one b64, conflict-free banks
        const v2f bm = *(const v2f*)(xs + (t * 16 + lm) * KSTR + kk + kb);
        acc[t] = WMMA_F32(a, bm, acc[t]);
      }
    }
  }

  // spill y accumulators, r-minor: ys[p][r] -> 2x ds_store_b128 per tile
#pragma unroll
  for (int t = 0; t < NT1; ++t) {
    int p = t * 16 + lm;
    if (p < NP) {
      float* yp = ys + p * RSTR + r0 + rowoff;
      *(v4f*)(yp)     = (v4f){acc[t][0], acc[t][1], acc[t][2], acc[t][3]};
      *(v4f*)(yp + 4) = (v4f){acc[t][4], acc[t][5], acc[t][6], acc[t][7]};
    }
  }
  __syncthreads();   // xs/f3t reads + ys writes complete before zs overwrite

  // ================= Phase 2: depthwise 3x3 (separable f1 (x) f2), y -> z
  {
    const int r    = tid >> 1;         // each rank handled by 2 threads
    const int half = tid & 1;
    const float wa0 = f1[0 * R_ + r], wa1 = f1[1 * R_ + r], wa2 = f1[2 * R_ + r];
    const float wb0 = f2[0 * R_ + r], wb1 = f2[1 * R_ + r], wb2 = f2[2 * R_ + r];
    const float w00 = wa0*wb0, w01 = wa0*wb1, w02 = wa0*wb2;
    const float w10 = wa1*wb0, w11 = wa1*wb1, w12 = wa1*wb2;
    const float w20 = wa2*wb0, w21 = wa2*wb1, w22 = wa2*wb2;
#pragma unroll 4
    for (int i = 0; i < 64; ++i) {
      int op   = (half << 6) + i;      // 128 output positions per rank
      int orow = op >> 4;
      int oc   = op & 15;
      const float* y0 = ys + (orow * YC + oc) * RSTR + r;   // tap (a,c) at +(a*YC+c)*RSTR
      float s = w00 * y0[0]               + w01 * y0[RSTR]             + w02 * y0[2*RSTR]
              + w10 * y0[YC*RSTR]         + w11 * y0[(YC+1)*RSTR]      + w12 * y0[(YC+2)*RSTR]
              + w20 * y0[2*YC*RSTR]       + w21 * y0[(2*YC+1)*RSTR]    + w22 * y0[(2*YC+2)*RSTR];
      zs[op * RSTR + r] = s;
    }
  }
  __syncthreads();

  // ================= Phase 3: out[f,op] = sum_r f0[f,r] * z[r,op]
  const int fb = wv << 5;              // this wave: 32 filters (2 M-tiles) x 8 N-tiles
  v8f oacc[2][8];
#pragma unroll
  for (int m = 0; m < 2; ++m)
#pragma unroll
    for (int t = 0; t < 8; ++t) oacc[m][t] = (v8f){0.f,0.f,0.f,0.f,0.f,0.f,0.f,0.f};

  const float* f0a = f0 + (size_t)(fb + lm) * R_;       // contiguous in r -> global b64
  const float* f0b = f0a + 16 * R_;

#pragma unroll 2
  for (int rk = 0; rk < R_; rk += 4) {
    const int rr = rk + kb;
    const v2f a0 = *(const v2f*)(f0a + rr);
    const v2f a1 = *(const v2f*)(f0b + rr);
#pragma unroll
    for (int t = 0; t < 8; ++t) {
      const v2f bm = *(const v2f*)(zs + (t * 16 + lm) * RSTR + rr);  // one b64, all 64 banks
      oacc[0][t] = WMMA_F32(a0, bm, oacc[0][t]);   // B reused across both M tiles
      oacc[1][t] = WMMA_F32(a1, bm, oacc[1][t]);
    }
  }

  // store: N-tile t == output row oh0+t, lanes 0-15 = consecutive cols (coalesced)
  float* ob = out + (size_t)b * F_ * HP_ * WP_;
#pragma unroll
  for (int m = 0; m < 2; ++m) {
#pragma unroll
    for (int t = 0; t < 8; ++t) {
      const int hp = oh0 + t;
      const int wp = ow0 + lm;
      if (hp < HP_ && wp < WP_) {
#pragma unroll
        for (int j = 0; j < 8; ++j) {
          const int f = fb + (m << 4) + rowoff + j;
          ob[((size_t)f * HP_ + hp) * WP_ + wp] = oacc[m][t][j];
        }
      }
    }
  }
}

extern "C" void kernel_launch(void* const* d_in, const int* in_sizes, int n_in,
                              void* d_out, int out_size, void* d_ws, size_t ws_size,
                              hipStream_t stream) {
  const float* x  = (const float*)d_in[0];
  const float* f0 = (const float*)d_in[1];
  const float* f1 = (const float*)d_in[2];
  const float* f2 = (const float*)d_in[3];
  const float* f3 = (const float*)d_in[4];
  float* out = (float*)d_out;

  dim3 grid(W_ / TOC, H_ / TOR, B_);   // (8, 16, 32) = 4096 workgroups
  dim3 block(256);                      // 8 wave32 waves
  size_t lds_bytes = (size_t)LDS_FLOATS * sizeof(float);  // ~159 KB of 320 KB WGP LDS
  hipLaunchKernelGGL(cp_conv_fused, grid, block, lds_bytes, stream,
                     x, f0, f1, f2, f3, out);
}